// DAGModel_88630945120510
// MI455X (gfx1250) — compile-verified
//
#include <hip/hip_runtime.h>

// DAG MLP for MI455X (gfx1250, wave32). fp32 WMMA path: V_WMMA_F32_16X16X4_F32.
//
// d_out doubles as the node-vector buffer: out row (s-1) <=> buffer slot s,
// slot 0 (padding) is skipped in the gather. 1 init kernel + 32 sequential
// depth kernels; stream ordering provides the inter-depth dependency.

typedef float v2f __attribute__((ext_vector_type(2)));
typedef float v8f __attribute__((ext_vector_type(8)));

#define BB   32      // batch
#define HH   128     // hidden
#define EE   64      // embedding
#define DD   32      // depths
#define NN   512     // nodes per depth
#define PP   8       // parents per node
#define FF   192     // H + E  (GEMM1 K)
#define ROWS 16385   // 1 + D*N  (output rows per batch element)

#define SXS  200     // LDS row stride of X tile (192 cols + pad, 200%64=8)
#define SHS  132     // LDS row stride of h tile (128 cols + pad, 132%64=4)

__global__ __launch_bounds__(256)
void dag_root_kernel(const float* __restrict__ embedding, float* __restrict__ out) {
    int i = blockIdx.x * 256 + threadIdx.x;          // over B*H = 4096
    if (i < BB * HH) {
        int b = i >> 7, h = i & 127;
        out[(size_t)b * ROWS * HH + h] = embedding[i];   // row 0 = root (slot 1)
    }
}

__global__ __launch_bounds__(256)
void dag_depth_kernel(const float* __restrict__ emb_table,
                      const float* __restrict__ W1, const float* __restrict__ b1,
                      const float* __restrict__ W2, const float* __restrict__ b2,
                      const int*   __restrict__ parent_idx,
                      float* __restrict__ out, int d) {
    __shared__ float sX[16 * SXS];   // X tile: [16 rows][192] (cols 0..127 = pv, 128..191 = node emb)
    __shared__ float sH[16 * SHS];   // h1 tile: [16 rows][128]
    __shared__ int   sPid[PP];

    const int tid = threadIdx.x;
    const int t   = blockIdx.x;      // 0..1023
    const int n   = t >> 1;          // node within depth
    const int b0  = (t & 1) << 4;    // batch base of this 16-row tile

    if (tid < PP) sPid[tid] = parent_idx[(d * NN + n) * PP + tid];
    __syncthreads();

    // ---- Phase 1a: gather+sum 8 parents -> pv in sX[:,0:128] ----
    {
        int r  = tid >> 4;            // 0..15 (batch row within tile)
        int c8 = (tid & 15) << 3;     // 0,8,...,120
        float acc[8];
#pragma unroll
        for (int j = 0; j < 8; ++j) acc[j] = 0.0f;
        size_t bbase = (size_t)(b0 + r) * ROWS * HH;
#pragma unroll
        for (int p = 0; p < PP; ++p) {
            int s = sPid[p];
            if (s > 0) {                               // slot 0 = zero padding row
                const float* src = out + bbase + (size_t)(s - 1) * HH + c8;
#pragma unroll
                for (int j = 0; j < 8; ++j) acc[j] += src[j];
            }
        }
#pragma unroll
        for (int j = 0; j < 8; ++j) sX[r * SXS + c8 + j] = acc[j];
    }
    // ---- Phase 1b: node embedding -> sX[:,128:192] (same row for all 16 b) ----
    if (tid < 128) {
        int r  = tid >> 3;
        int c8 = (tid & 7) << 3;      // 0..56
        const float* src = emb_table + (size_t)(2 + d * NN + n) * EE + c8;
#pragma unroll
        for (int j = 0; j < 8; ++j) sX[r * SXS + HH + c8 + j] = src[j];
    }
    __syncthreads();

    const int lane  = tid & 31;
    const int wave  = tid >> 5;            // 0..7 -> 16-wide output column tile
    const int obase = wave << 4;
    const int col   = lane & 15;           // A-row / B-col / C-col index
    const int hi    = lane >> 4;           // lane half selects K pair / M half
    const int khalf = hi << 1;             // 0 or 2

    // ---- GEMM1: h1 = relu(X @ W1^T + b1), K = 192, 48 chained f32 WMMAs ----
    v8f acc;
    {
        float bv = b1[obase + col];
#pragma unroll
        for (int v = 0; v < 8; ++v) acc[v] = bv;   // bias pre-loaded into C
    }
    const float* w1row = W1 + (size_t)(obase + col) * FF + khalf;  // B[k][o] = W1[o][k]
#pragma unroll 4
    for (int k0 = 0; k0 < FF; k0 += 4) {
        v2f a, bf;
        a[0]  = sX[col * SXS + k0 + khalf];
        a[1]  = sX[col * SXS + k0 + khalf + 1];
        bf[0] = w1row[k0];
        bf[1] = w1row[k0 + 1];
        acc = __builtin_amdgcn_wmma_f32_16x16x4_f32(false, a, false, bf,
                                                    (short)0, acc, false, false);
    }
    // ReLU, write h1 tile to LDS
#pragma unroll
    for (int v = 0; v < 8; ++v) {
        int m = v + (hi << 3);                       // C/D: M = v + 8*(lane>>4)
        sH[m * SHS + obase + col] = acc[v] > 0.0f ? acc[v] : 0.0f;
    }
    __syncthreads();

    // ---- GEMM2: y = h1 @ W2^T + b2, K = 128, 32 chained f32 WMMAs ----
    v8f acc2;
    {
        float bv = b2[obase + col];
#pragma unroll
        for (int v = 0; v < 8; ++v) acc2[v] = bv;
    }
    const float* w2row = W2 + (size_t)(obase + col) * HH + khalf;
#pragma unroll 4
    for (int k0 = 0; k0 < HH; k0 += 4) {
        v2f a, bf;
        a[0]  = sH[col * SHS + k0 + khalf];
        a[1]  = sH[col * SHS + k0 + khalf + 1];
        bf[0] = w2row[k0];
        bf[1] = w2row[k0 + 1];
        acc2 = __builtin_amdgcn_wmma_f32_16x16x4_f32(false, a, false, bf,
                                                     (short)0, acc2, false, false);
    }

    // ---- residual (pv lives in sX[:,0:128]) + store to out row 1 + d*N + n ----
    const int orow = 1 + d * NN + n;
#pragma unroll
    for (int v = 0; v < 8; ++v) {
        int m = v + (hi << 3);
        float val = acc2[v] + sX[m * SXS + obase + col];
        out[((size_t)(b0 + m) * ROWS + orow) * HH + obase + col] = val;
    }
}

extern "C" void kernel_launch(void* const* d_in, const int* in_sizes, int n_in,
                              void* d_out, int out_size, void* d_ws, size_t ws_size,
                              hipStream_t stream) {
    const float* embedding  = (const float*)d_in[0];
    const float* emb_table  = (const float*)d_in[1];
    const float* W1         = (const float*)d_in[2];
    const float* b1         = (const float*)d_in[3];
    const float* W2         = (const float*)d_in[4];
    const float* b2         = (const float*)d_in[5];
    const int*   parent_idx = (const int*)d_in[6];
    float* out = (float*)d_out;

    dag_root_kernel<<<(BB * HH + 255) / 256, 256, 0, stream>>>(embedding, out);

    // 32 level-synchronous steps; stream order = inter-depth barrier.
    for (int d = 0; d < DD; ++d) {
        dag_depth_kernel<<<(BB * NN) / 16, 256, 0, stream>>>(
            emb_table, W1, b1, W2, b2, parent_idx, out, d);
    }
}